// LengthTTVLossLayer_1623497638139
// MI455X (gfx1250) — compile-verified
//
#include <hip/hip_runtime.h>
#include <cstdint>
#include <cstddef>

// ---------------------------------------------------------------------------
// LengthTTVLoss on gfx1250 (MI455X).
//   cam_output: (B=16, T=1024, H=14, W=14, C=16) f32  ~205MB -> memory bound
//   (23.3 TB/s => ~8.8us floor). Pass 1: one wave per (b,t) tile; channel
//   sums via V_WMMA_F32_16X16X4_F32 (A = ones, 8 positions per issue); KL
//   log-sum accumulated in log2 domain (raw v_log_f32), converted by ln2
//   once per wave. Pass 2: single block, fixed-order reduction (no atomics,
//   bitwise deterministic across graph replays).
// ---------------------------------------------------------------------------

typedef __attribute__((ext_vector_type(2))) float v2f;
typedef __attribute__((ext_vector_type(8))) float v8f;

#define B_      16
#define T_      1024
#define HW_     196
#define C_      16
#define EPS_    1e-7f
#define TCHUNK  8                         // t's per block (one per wave)
#define BLKS_PER_B (T_ / TCHUNK)          // 128
#define NBLK    (B_ * BLKS_PER_B)         // 2048
#define PART_STRIDE 33                    // 16 active + 16 blank + 1 kl

#define LN2_        0.6931471805599453f
#define LOG_INV_H_  (-2.6390573296152584f)   // ln(1/14)

__device__ __forceinline__ float clamp01e(float v) {
    return fminf(fmaxf(v, EPS_), 1.0f);      // -> v_med3_num_f32
}

__global__ __launch_bounds__(256) void ttv_partial_kernel(
    const float* __restrict__ cam,
    const long long* __restrict__ length,
    float* __restrict__ part)
{
    const int tid  = threadIdx.x;
    const int wave = tid >> 5;
    const int lane = tid & 31;
    const int half = lane >> 4;           // which position within a pair
    const int c    = lane & 15;           // channel owned by this lane

    const int b = blockIdx.x / BLKS_PER_B;
    const int t = (blockIdx.x % BLKS_PER_B) * TCHUNK + wave;

    const bool m = ((long long)t < length[b]);          // wave-uniform mask

    const float* base = cam + ((size_t)b * T_ + t) * (size_t)(HW_ * C_)
                            + (size_t)half * C_ + c;

    // A matrix (16x4, all ones): every (lane, vgpr) slot is 1.0f.
    v2f a; a.x = 1.0f; a.y = 1.0f;
    v8f acc = {};
    float l2sum = 0.0f;   // sum of log2(clamp(v)) over this lane's elements

    if (m) {
        // ---- active tile: stream + WMMA channel sums + log2 accumulation
        for (int p = 0; p < HW_ - 4; p += 8) {
            float v0 = __builtin_nontemporal_load(base + (p    ) * C_);
            float v1 = __builtin_nontemporal_load(base + (p + 2) * C_);
            float v2 = __builtin_nontemporal_load(base + (p + 4) * C_);
            float v3 = __builtin_nontemporal_load(base + (p + 6) * C_);
            v2f bm; bm.x = v0 + v1; bm.y = v2 + v3;     // 8 positions / WMMA
            acc = __builtin_amdgcn_wmma_f32_16x16x4_f32(
                false, a, false, bm, (short)0, acc, false, false);
            l2sum += __builtin_amdgcn_logf(clamp01e(v0));
            l2sum += __builtin_amdgcn_logf(clamp01e(v1));
            l2sum += __builtin_amdgcn_logf(clamp01e(v2));
            l2sum += __builtin_amdgcn_logf(clamp01e(v3));
        }
        {   // tail: positions 192..195
            float v0 = __builtin_nontemporal_load(base + 192 * C_);
            float v1 = __builtin_nontemporal_load(base + 194 * C_);
            v2f bm; bm.x = v0; bm.y = v1;
            acc = __builtin_amdgcn_wmma_f32_16x16x4_f32(
                false, a, false, bm, (short)0, acc, false, false);
            l2sum += __builtin_amdgcn_logf(clamp01e(v0));
            l2sum += __builtin_amdgcn_logf(clamp01e(v1));
        }
    } else {
        // ---- blank tile: pure streaming reduction, no transcendental work
        for (int p = 0; p < HW_ - 4; p += 8) {
            float v0 = __builtin_nontemporal_load(base + (p    ) * C_);
            float v1 = __builtin_nontemporal_load(base + (p + 2) * C_);
            float v2 = __builtin_nontemporal_load(base + (p + 4) * C_);
            float v3 = __builtin_nontemporal_load(base + (p + 6) * C_);
            v2f bm; bm.x = v0 + v1; bm.y = v2 + v3;
            acc = __builtin_amdgcn_wmma_f32_16x16x4_f32(
                false, a, false, bm, (short)0, acc, false, false);
        }
        {
            float v0 = __builtin_nontemporal_load(base + 192 * C_);
            float v1 = __builtin_nontemporal_load(base + 194 * C_);
            v2f bm; bm.x = v0; bm.y = v1;
            acc = __builtin_amdgcn_wmma_f32_16x16x4_f32(
                false, a, false, bm, (short)0, acc, false, false);
        }
    }

    // Reduce log2-sum across the wave (fixed order -> deterministic).
    for (int off = 16; off > 0; off >>= 1)
        l2sum += __shfl_xor(l2sum, off, 32);

    // KL contribution of this (b,t) tile (blank tiles: eps*log(1) == 0).
    const float inv_h = 1.0f / 14.0f;
    float klp = m ? inv_h * ((float)(HW_ * C_) * LOG_INV_H_ - LN2_ * l2sum)
                  : 0.0f;

    // Channel sum for channel (lane&15) sits in acc[0] (D row M=0 / M=8).
    float chsum = acc[0];

    __shared__ float lds_a[TCHUNK][C_];
    __shared__ float lds_b[TCHUNK][C_];
    __shared__ float lds_k[TCHUNK];
    if (lane < 16) {
        lds_a[wave][lane] = m ? chsum : 0.0f;
        lds_b[wave][lane] = m ? 0.0f  : chsum;
    }
    if (lane == 0) lds_k[wave] = klp;
    __syncthreads();

    // Fixed-order cross-wave reduction; write block partial (no atomics).
    float* outp = part + (size_t)blockIdx.x * PART_STRIDE;
    if (tid < 16) {
        float s = 0.0f;
        for (int w = 0; w < TCHUNK; ++w) s += lds_a[w][tid];
        outp[tid] = s;
    } else if (tid < 32) {
        float s = 0.0f; const int cc = tid - 16;
        for (int w = 0; w < TCHUNK; ++w) s += lds_b[w][cc];
        outp[16 + cc] = s;
    } else if (tid == 32) {
        float s = 0.0f;
        for (int w = 0; w < TCHUNK; ++w) s += lds_k[w];
        outp[32] = s;
    }
}

__global__ __launch_bounds__(256) void ttv_final_kernel(
    const float* __restrict__ part,
    const float* __restrict__ count,
    float* __restrict__ out)
{
    const int tid = threadIdx.x;
    const int b = tid >> 4;
    const int c = tid & 15;

    // Fixed-order accumulation of the 128 partials belonging to batch b.
    float asum = 0.0f, bsum = 0.0f;
    for (int j = 0; j < BLKS_PER_B; ++j) {
        const float* p = part + (size_t)(b * BLKS_PER_B + j) * PART_STRIDE;
        asum += p[c];
        bsum += p[16 + c];
    }

    // Huber (delta = 1): 0.5*quad^2 + (err - quad)
    float ea = fabsf(asum - count[b * C_ + c]);
    float qa = fminf(ea, 1.0f);
    float ha = 0.5f * qa * qa + (ea - qa);
    float eb = fabsf(bsum);
    float qb = fminf(eb, 1.0f);
    float hb = 0.5f * qb * qb + (eb - qb);

    __shared__ float s_h[256];
    __shared__ float s_kl[256];
    s_h[tid] = ha + hb;

    float kp = 0.0f;
    for (int j = tid; j < NBLK; j += 256)
        kp += part[(size_t)j * PART_STRIDE + 32];
    s_kl[tid] = kp;
    __syncthreads();

    for (int off = 128; off > 0; off >>= 1) {
        if (tid < off) s_kl[tid] += s_kl[tid + off];
        __syncthreads();
    }

    if (c == 0) {
        float hm = 0.0f;
        for (int cc = 0; cc < C_; ++cc) hm += s_h[(b << 4) + cc];
        hm *= (1.0f / 16.0f);                     // mean over channels
        float kl_loss = 0.1f * s_kl[0] / ((float)B_ * (float)T_ * (float)HW_);
        out[b] = hm + kl_loss;                    // active_w = blank_w = 1
    }
}

extern "C" void kernel_launch(void* const* d_in, const int* in_sizes, int n_in,
                              void* d_out, int out_size, void* d_ws, size_t ws_size,
                              hipStream_t stream) {
    const float*      cam    = (const float*)d_in[0];
    const float*      count  = (const float*)d_in[1];
    const long long*  length = (const long long*)d_in[2];   // int64 in reference
    float*            out    = (float*)d_out;
    float*            part   = (float*)d_ws;                // needs NBLK*33*4 ~ 270 KB

    ttv_partial_kernel<<<NBLK, 256, 0, stream>>>(cam, length, part);
    ttv_final_kernel<<<1, 256, 0, stream>>>(part, count, out);
}